// GCNLayer_pyg_40785009443358
// MI455X (gfx1250) — compile-verified
//
#include <hip/hip_runtime.h>
#include <hip/hip_bf16.h>
#include <cstdint>
#include <cstddef>

// CDNA5 wave32 WMMA fragment types
typedef __attribute__((ext_vector_type(2))) float v2f;  // f32 A/B frag for 16x16x4
typedef __attribute__((ext_vector_type(8))) float v8f;  // f32 C/D frag (16x16)

#define DFEAT 128

// ---------------- degree / normalization ----------------
__global__ void k_deg_init(float* __restrict__ deg, int N) {
  int i = blockIdx.x * blockDim.x + threadIdx.x;
  if (i < N) deg[i] = 1.0f;  // self-loop contributes 1 to in-degree
}

__global__ void k_zero(float* __restrict__ p, int n) {
  int i = blockIdx.x * blockDim.x + threadIdx.x;
  if (i < n) p[i] = 0.0f;
}

__global__ void k_deg_accum(const int* __restrict__ dst, float* __restrict__ deg, int E) {
  int e = blockIdx.x * blockDim.x + threadIdx.x;
  if (e < E) atomicAdd(&deg[dst[e]], 1.0f);
}

__global__ void k_deg_rsqrt(float* __restrict__ deg, int N) {
  int i = blockIdx.x * blockDim.x + threadIdx.x;
  if (i < N) deg[i] = rsqrtf(deg[i]);  // deg >= 1 always (self-loop), no zero guard needed
}

// ---------------- fused dual GEMM via fp32 WMMA ----------------
// Block = 128 threads = 4 waves. Block covers 16 rows; wave w covers cols [32w, 32w+32).
// Both GEMMs (x@W and x@res_W) share the A fragment in a single K loop.
// Layout per ISA 7.12.2 (f32):
//   A 16x4:  lane L, vgpr j  -> A[M = L&15][K = j + 2*(L>>4)]
//   B 4x16:  lane L, vgpr j  -> B[K = j + 2*(L>>4)][N = L&15]
//   C 16x16: lane L, vgpr j  -> C[M = j + 8*(L>>4)][N = L&15]
__global__ void __launch_bounds__(128)
k_gemm_fused(const float* __restrict__ x,
             const float* __restrict__ W,
             const float* __restrict__ bias,
             const float* __restrict__ resW,
             const float* __restrict__ resb,
             const float* __restrict__ dis,
             float* __restrict__ hs,    // [N,128]  h * dis[row]
             float* __restrict__ y,     // [N,128]  b + dis^2*h + relu(x@resW + resb)
             int N)
{
  const int wave  = threadIdx.x >> 5;
  const int lane  = threadIdx.x & 31;
  const int lhalf = lane >> 4;   // 0 or 1
  const int lmod  = lane & 15;
  const int rowBase = blockIdx.x * 16;
  if (rowBase >= N) return;      // wave-uniform: EXEC stays all-1s for WMMA
  const int colBase = wave * 32;

  const float* xrow = x + (size_t)(rowBase + lmod) * DFEAT;

  v8f accW[2] = {};
  v8f accR[2] = {};

  #pragma unroll 4
  for (int k = 0; k < 32; ++k) {
    const int kb = k * 4 + 2 * lhalf;           // even -> 8B aligned
    const v2f a = *(const v2f*)(xrow + kb);     // A frag: 2 consecutive K values
    #pragma unroll
    for (int n = 0; n < 2; ++n) {
      const int col = colBase + n * 16 + lmod;
      v2f bw, br;
      bw.x = W[(size_t)kb * DFEAT + col];
      bw.y = W[(size_t)(kb + 1) * DFEAT + col];
      br.x = resW[(size_t)kb * DFEAT + col];
      br.y = resW[(size_t)(kb + 1) * DFEAT + col];
      accW[n] = __builtin_amdgcn_wmma_f32_16x16x4_f32(
          false, a, false, bw, (short)0, accW[n], false, false);
      accR[n] = __builtin_amdgcn_wmma_f32_16x16x4_f32(
          false, a, false, br, (short)0, accR[n], false, false);
    }
  }

  #pragma unroll
  for (int n = 0; n < 2; ++n) {
    const int col   = colBase + n * 16 + lmod;
    const float bc  = bias[col];
    const float rbc = resb[col];
    #pragma unroll
    for (int j = 0; j < 8; ++j) {
      const int row  = rowBase + j + 8 * lhalf;
      const float d  = dis[row];
      const float hv = accW[n][j];
      const float hsv = hv * d;
      hs[(size_t)row * DFEAT + col] = hsv;
      const float rv = fmaxf(accR[n][j] + rbc, 0.0f);
      y[(size_t)row * DFEAT + col] = bc + d * hsv + rv;  // bias + self-loop msg + residual
    }
  }
}

// ---------------- edge scatter: y[dst] += hs[src] * dis[dst] ----------------
__global__ void __launch_bounds__(256)
k_scatter(const int* __restrict__ src, const int* __restrict__ dst,
          const float* __restrict__ dis, const float* __restrict__ hs,
          float* __restrict__ y, int E)
{
  const int e = blockIdx.x * 8 + (threadIdx.x >> 5);
  if (e >= E) return;
  const int lane = threadIdx.x & 31;
  const int s = src[e];
  const int d = dst[e];
  const float nd = dis[d];
  const float4 v = *(const float4*)(hs + (size_t)s * DFEAT + lane * 4);
  float* yp = y + (size_t)d * DFEAT + lane * 4;
  atomicAdd(yp + 0, v.x * nd);
  atomicAdd(yp + 1, v.y * nd);
  atomicAdd(yp + 2, v.z * nd);
  atomicAdd(yp + 3, v.w * nd);
}

// ---------------- BatchNorm (batch stats, biased variance) ----------------
__global__ void __launch_bounds__(128)
k_bn_stats(const float* __restrict__ y, float* __restrict__ sums,
           float* __restrict__ sumsq, int N)
{
  const int col = threadIdx.x;  // 128 threads = 128 features, coalesced row reads
  float s = 0.0f, ss = 0.0f;
  for (int r = blockIdx.x; r < N; r += gridDim.x) {
    const float v = y[(size_t)r * DFEAT + col];
    s += v;
    ss += v * v;
  }
  atomicAdd(&sums[col], s);
  atomicAdd(&sumsq[col], ss);
}

__global__ void k_bn_finalize(const float* __restrict__ sums, const float* __restrict__ sumsq,
                              const float* __restrict__ gamma, const float* __restrict__ beta,
                              float* __restrict__ scale, float* __restrict__ shift, int N)
{
  const int f = threadIdx.x;
  if (f < DFEAT) {
    const float invN = 1.0f / (float)N;
    const float m    = sums[f] * invN;
    const float var  = fmaxf(sumsq[f] * invN - m * m, 0.0f);  // biased variance
    const float sc   = gamma[f] * rsqrtf(var + 1e-5f);
    scale[f] = sc;
    shift[f] = beta[f] - m * sc;
  }
}

__global__ void __launch_bounds__(256)
k_bn_apply(float* __restrict__ y, const float* __restrict__ scale,
           const float* __restrict__ shift, long n4)
{
  const long i = (long)blockIdx.x * blockDim.x + threadIdx.x;
  if (i >= n4) return;
  const int c = (int)(i & 31) * 4;  // 32 float4 per 128-wide row
  float4 v = ((const float4*)y)[i];
  v.x = v.x * scale[c + 0] + shift[c + 0];
  v.y = v.y * scale[c + 1] + shift[c + 1];
  v.z = v.z * scale[c + 2] + shift[c + 2];
  v.w = v.w * scale[c + 3] + shift[c + 3];
  ((float4*)y)[i] = v;
}

// ---------------- host launch ----------------
extern "C" void kernel_launch(void* const* d_in, const int* in_sizes, int n_in,
                              void* d_out, int out_size, void* d_ws, size_t ws_size,
                              hipStream_t stream)
{
  const float* x     = (const float*)d_in[0];
  const int*   ei    = (const int*)d_in[1];  // int32: JAX x64 disabled by default
  const float* W     = (const float*)d_in[2];
  const float* bias  = (const float*)d_in[3];
  const float* resW  = (const float*)d_in[4];
  const float* resb  = (const float*)d_in[5];
  const float* gamma = (const float*)d_in[6];
  const float* beta  = (const float*)d_in[7];
  float* y = (float*)d_out;

  const int N = in_sizes[0] / DFEAT;
  const int E = in_sizes[1] / 2;
  const int* src = ei;       // edge_index[0]
  const int* dst = ei + E;   // edge_index[1]

  // workspace layout (floats): hs[N*128] | deg/dis[N] | sums[128] | sumsq[128] | scale[128] | shift[128]
  float* hs    = (float*)d_ws;
  float* deg   = hs + (size_t)N * DFEAT;
  float* sums  = deg + N;
  float* sumsq = sums + DFEAT;
  float* scale = sumsq + DFEAT;
  float* shift = scale + DFEAT;

  k_deg_init<<<(N + 255) / 256, 256, 0, stream>>>(deg, N);
  k_zero<<<2, 256, 0, stream>>>(sums, 4 * DFEAT);
  k_deg_accum<<<(E + 255) / 256, 256, 0, stream>>>(dst, deg, E);
  k_deg_rsqrt<<<(N + 255) / 256, 256, 0, stream>>>(deg, N);

  k_gemm_fused<<<(N + 15) / 16, 128, 0, stream>>>(x, W, bias, resW, resb, deg, hs, y, N);

  k_scatter<<<(E + 7) / 8, 256, 0, stream>>>(src, dst, deg, hs, y, E);

  k_bn_stats<<<512, 128, 0, stream>>>(y, sums, sumsq, N);
  k_bn_finalize<<<1, 128, 0, stream>>>(sums, sumsq, gamma, beta, scale, shift, N);
  const long n4 = (long)N * (DFEAT / 4);
  k_bn_apply<<<(int)((n4 + 255) / 256), 256, 0, stream>>>(y, scale, shift, n4);
}